// Classifer_SelectSuperparameter_22299470201683
// MI455X (gfx1250) — compile-verified
//
#include <hip/hip_runtime.h>
#include <math.h>

// ---------------------------------------------------------------------------
// Problem constants (fixed by the reference)
// ---------------------------------------------------------------------------
#define NN    50000     // 3125 * 16
#define EE    800000
#define GG    2000      // 125 * 16
#define F_IN  38
#define F_INP 40        // padded K for layer-1 GEMM (mult of 8)
#define HID   128
#define M0D   500
#define M0P   512
#define M1D   100
#define M1P   112
#define OUTD  67
#define OUTP  80
#define EPS   1e-5f

typedef __attribute__((ext_vector_type(2))) float v2f;
typedef __attribute__((ext_vector_type(8))) float v8f;

// ---------------------------------------------------------------------------
// Utility kernels
// ---------------------------------------------------------------------------
__global__ void fill0_kernel(float* __restrict__ p, int n) {
    for (int i = blockIdx.x * blockDim.x + threadIdx.x; i < n;
         i += gridDim.x * blockDim.x)
        p[i] = 0.0f;
}

// dst[KP x NP] = zero-padded src[K x N]
__global__ void pad2d_kernel(const float* __restrict__ src, float* __restrict__ dst,
                             int K, int N, int KP, int NP) {
    const int total = KP * NP;
    for (int i = blockIdx.x * blockDim.x + threadIdx.x; i < total;
         i += gridDim.x * blockDim.x) {
        int r = i / NP, c = i - r * NP;
        dst[i] = (r < K && c < N) ? src[r * N + c] : 0.0f;
    }
}

__global__ void degree_kernel(const int* __restrict__ src, const int* __restrict__ dst,
                              float* __restrict__ degO, float* __restrict__ degI, int E) {
    for (int e = blockIdx.x * blockDim.x + threadIdx.x; e < E;
         e += gridDim.x * blockDim.x) {
        atomicAdd(&degO[src[e]], 1.0f);
        atomicAdd(&degI[dst[e]], 1.0f);
    }
}

// deg -> rsqrt(max(deg,1)) in place
__global__ void norm_kernel(float* __restrict__ p, int n) {
    for (int i = blockIdx.x * blockDim.x + threadIdx.x; i < n;
         i += gridDim.x * blockDim.x)
        p[i] = rsqrtf(fmaxf(p[i], 1.0f));
}

// agg[dst[e]*ldA + f] += h[src[e]*ldH + f] * normOut[src[e]]   for f in [0,F)
__global__ void scatter_kernel(const float* __restrict__ h, const int* __restrict__ src,
                               const int* __restrict__ dst, const float* __restrict__ normOut,
                               float* __restrict__ agg, int total, int F, int ldH, int ldA) {
    for (int i = blockIdx.x * blockDim.x + threadIdx.x; i < total;
         i += gridDim.x * blockDim.x) {
        int e = i / F, f = i - e * F;
        int s = src[e], d = dst[e];
        atomicAdd(&agg[(long long)d * ldA + f], h[(long long)s * ldH + f] * normOut[s]);
    }
}

// ---------------------------------------------------------------------------
// Aligned WMMA fp32 GEMM:
//   C[M,Nc] = diag(rowScale?) * A[M,K] @ B[K,Nc] + bias
// Requirements (guaranteed by caller via padding): M%16==0, K%8==0, Nc%16==0.
// One wave computes one 16x16 tile via V_WMMA_F32_16X16X4_F32; no guards in
// the inner loop -> straight-line load/wmma body.
// A frag (16x4 f32): lane l<16 holds A[row0+l][k..k+1], lanes 16..31 hold
// K {k+2,k+3}; the two K values are adjacent -> single b64 load.
// C/D: vgpr i, lane -> row = row0 + i + 8*(lane>=16), col = col0 + (lane&15).
// ---------------------------------------------------------------------------
__global__ void gemm_wmma_f32(const float* __restrict__ A, const float* __restrict__ B,
                              const float* __restrict__ bias, const float* __restrict__ rowScale,
                              float* __restrict__ C, int M, int K, int Nc) {
    const int wave   = threadIdx.x >> 5;
    const int lane   = threadIdx.x & 31;
    const int l      = lane & 15;
    const int half   = lane >> 4;
    const int tilesN = Nc >> 4;
    const int tile   = blockIdx.x * (blockDim.x >> 5) + wave;
    if (tile >= (M >> 4) * tilesN) return;

    const int tm   = tile / tilesN;
    const int tn   = tile - tm * tilesN;
    const int row0 = tm << 4;
    const int col0 = tn << 4;
    const int arow = row0 + l;
    const int bcol = col0 + l;

    const float ascl = (rowScale != nullptr) ? rowScale[arow] : 1.0f;
    const float bb   = (bias != nullptr) ? bias[bcol] : 0.0f;
    v8f c = (v8f){bb, bb, bb, bb, bb, bb, bb, bb};

    const float* ap = A + (long long)arow * K + (half << 1);
    const float* bp = B + (long long)(half << 1) * Nc + bcol;
    const long long bstep = (long long)Nc;

    for (int k = 0; k < K; k += 8) {
        v2f a0 = *(const v2f*)(ap);            // A[arow][k+2h .. k+2h+1] (b64)
        v2f b0;
        b0.x = bp[0];                           // B[k+2h  ][bcol]
        b0.y = bp[bstep];                       // B[k+2h+1][bcol]
        a0.x *= ascl; a0.y *= ascl;
        c = __builtin_amdgcn_wmma_f32_16x16x4_f32(false, a0, false, b0,
                                                  (short)0, c, false, false);
        v2f a1 = *(const v2f*)(ap + 4);
        v2f b1;
        b1.x = bp[4 * bstep];
        b1.y = bp[5 * bstep];
        a1.x *= ascl; a1.y *= ascl;
        c = __builtin_amdgcn_wmma_f32_16x16x4_f32(false, a1, false, b1,
                                                  (short)0, c, false, false);
        ap += 8;
        bp += 8 * bstep;
    }

    float* cp = C + (long long)(row0 + (half << 3)) * Nc + bcol;
#pragma unroll
    for (int i = 0; i < 8; ++i) {
        *cp = c[i];
        cp += Nc;
    }
}

// ---------------------------------------------------------------------------
// BatchNorm (training-mode batch statistics), two pass
// ---------------------------------------------------------------------------
__global__ void bn_reduce_kernel(const float* __restrict__ X, float* __restrict__ sums,
                                 float* __restrict__ sumsq, int M, int C) {
    const int c   = blockIdx.x;           // one block per channel
    const int tid = threadIdx.x;
    __shared__ float ss[256];
    __shared__ float sq[256];
    float a = 0.0f, b = 0.0f;
    for (int r = tid; r < M; r += blockDim.x) {
        float v = X[(long long)r * C + c];
        a += v; b += v * v;
    }
    ss[tid] = a; sq[tid] = b;
    __syncthreads();
    for (int s = 128; s > 0; s >>= 1) {
        if (tid < s) { ss[tid] += ss[tid + s]; sq[tid] += sq[tid + s]; }
        __syncthreads();
    }
    if (tid == 0) { sums[c] = ss[0]; sumsq[c] = sq[0]; }
}

__global__ void bn_apply_relu_kernel(float* __restrict__ X, const float* __restrict__ sums,
                                     const float* __restrict__ sumsq,
                                     const float* __restrict__ g, const float* __restrict__ b,
                                     int M, int C) {
    const int total = M * C;
    const float invM = 1.0f / (float)M;
    for (int i = blockIdx.x * blockDim.x + threadIdx.x; i < total;
         i += gridDim.x * blockDim.x) {
        int c = i % C;
        float m = sums[c] * invM;
        float v = sumsq[c] * invM - m * m;
        float y = (X[i] - m) * rsqrtf(v + EPS) * g[c] + b[c];
        X[i] = fmaxf(y, 0.0f);
    }
}

// ---------------------------------------------------------------------------
// Readout
// ---------------------------------------------------------------------------
__global__ void atom_weight_kernel(const float* __restrict__ h, const float* __restrict__ rwW,
                                   const float* __restrict__ rwb, float* __restrict__ awOut,
                                   float* __restrict__ sigw, int N, int H) {
    for (int n = blockIdx.x * blockDim.x + threadIdx.x; n < N;
         n += gridDim.x * blockDim.x) {
        float acc = 0.0f;
        const float* row = h + (long long)n * H;
        for (int k = 0; k < H; ++k) acc += row[k] * rwW[k];
        acc += rwb[0];
        awOut[n] = acc;
        sigw[n]  = 1.0f / (1.0f + expf(-acc));
    }
}

__global__ void readout_scatter_kernel(const float* __restrict__ h, const float* __restrict__ sigw,
                                       const int* __restrict__ gid, float* __restrict__ hg,
                                       int N, int H) {
    const int total = N * H;
    for (int i = blockIdx.x * blockDim.x + threadIdx.x; i < total;
         i += gridDim.x * blockDim.x) {
        int n = i / H, f = i - n * H;
        atomicAdd(&hg[(long long)gid[n] * H + f], h[i] * sigw[n]);
    }
}

// out[g*C + c] = sigmoid(X[g*CP + c])   (slice padded cols away)
__global__ void sigmoid_slice_kernel(const float* __restrict__ x, float* __restrict__ out,
                                     int G, int C, int CP) {
    const int total = G * C;
    for (int i = blockIdx.x * blockDim.x + threadIdx.x; i < total;
         i += gridDim.x * blockDim.x) {
        int gg = i / C, cc = i - gg * C;
        out[i] = 1.0f / (1.0f + expf(-x[(long long)gg * CP + cc]));
    }
}

// ---------------------------------------------------------------------------
// Host-side orchestration
// ---------------------------------------------------------------------------
static inline int cdiv(int a, int b) { return (a + b - 1) / b; }

extern "C" void kernel_launch(void* const* d_in, const int* in_sizes, int n_in,
                              void* d_out, int out_size, void* d_ws, size_t ws_size,
                              hipStream_t stream) {
    // Input order = setup_inputs() insertion order (params depth-first):
    const float* h_in    = (const float*)d_in[0];                     // [N,38]
    const float* gcnW[3] = { (const float*)d_in[1], (const float*)d_in[2], (const float*)d_in[3] };
    const float* gcnB[3] = { (const float*)d_in[4], (const float*)d_in[5], (const float*)d_in[6] };
    const float* gcnG[3] = { (const float*)d_in[7], (const float*)d_in[8], (const float*)d_in[9] };
    const float* gcnBeta[3] = { (const float*)d_in[10], (const float*)d_in[11], (const float*)d_in[12] };
    const float* rwW   = (const float*)d_in[13];
    const float* rwB   = (const float*)d_in[14];
    const float* fc1W  = (const float*)d_in[15];
    const float* fc1B  = (const float*)d_in[16];
    const float* bn1G  = (const float*)d_in[17];
    const float* bn1B  = (const float*)d_in[18];
    const float* linW  = (const float*)d_in[19];
    const float* linB  = (const float*)d_in[20];
    const float* bnG   = (const float*)d_in[21];
    const float* bnB   = (const float*)d_in[22];
    const float* fc2W  = (const float*)d_in[23];
    const float* fc2B  = (const float*)d_in[24];
    const int*   src   = (const int*)d_in[25];
    const int*   dst   = (const int*)d_in[26];
    const int*   gid   = (const int*)d_in[27];

    float* out   = (float*)d_out;               // [G*67] preds ++ [N] atom logits
    float* outAW = out + GG * OUTD;

    // Workspace carve-up (floats); all offsets even -> 8B alignment preserved
    float* ws = (float*)d_ws;
    float* normOut = ws;   ws += NN;                         // also deg_out
    float* normIn  = ws;   ws += NN;                         // also deg_in
    float* buf0    = ws;   ws += (long long)NN * HID;        // node features
    float* buf1    = ws;   ws += (long long)NN * HID;        // agg scratch (layer1 uses [N,40])
    float* sigw    = ws;   ws += NN;
    float* hg      = ws;   ws += (long long)GG * HID;
    float* mbuf0   = ws;   ws += (long long)GG * M0P;        // [G,512]
    float* mbuf1   = ws;   ws += (long long)GG * M1P;        // [G,112]
    float* mbuf2   = ws;   ws += (long long)GG * OUTP;       // [G,80]
    float* sums    = ws;   ws += M0P;
    float* sumsq   = ws;   ws += M0P;
    // padded parameter copies
    float* W0p     = ws;   ws += F_INP * HID;                // [40,128]
    float* fc1Wp   = ws;   ws += HID * M0P;                  // [128,512]
    float* fc1Bp   = ws;   ws += M0P;
    float* bn1Gp   = ws;   ws += M0P;
    float* bn1Bp   = ws;   ws += M0P;
    float* linWp   = ws;   ws += M0P * M1P;                  // [512,112]
    float* linBp   = ws;   ws += M1P;
    float* bnGp    = ws;   ws += M1P;
    float* bnBp    = ws;   ws += M1P;
    float* fc2Wp   = ws;   ws += M1P * OUTP;                 // [112,80]
    float* fc2Bp   = ws;   ws += OUTP;

    const int T = 256;
    auto pad = [&](const float* s, float* d, int K, int N, int KP, int NP) {
        pad2d_kernel<<<cdiv(KP * NP, T), T, 0, stream>>>(s, d, K, N, KP, NP);
    };
    auto gemm = [&](const float* A, const float* B, const float* bias,
                    const float* rs, float* C, int M, int K, int Nc) {
        int tiles = (M / 16) * (Nc / 16);
        gemm_wmma_f32<<<cdiv(tiles, 8), 256, 0, stream>>>(A, B, bias, rs, C, M, K, Nc);
    };
    auto bn = [&](float* X, const float* g, const float* b, int M, int C) {
        bn_reduce_kernel<<<C, 256, 0, stream>>>(X, sums, sumsq, M, C);
        bn_apply_relu_kernel<<<cdiv(M * C, T), T, 0, stream>>>(X, sums, sumsq, g, b, M, C);
    };

    // --- zero-padded parameter staging -------------------------------------
    pad(gcnW[0], W0p, F_IN, HID, F_INP, HID);
    pad(fc1W,  fc1Wp, HID, M0D, HID, M0P);
    pad(fc1B,  fc1Bp, 1, M0D, 1, M0P);
    pad(bn1G,  bn1Gp, 1, M0D, 1, M0P);
    pad(bn1B,  bn1Bp, 1, M0D, 1, M0P);
    pad(linW,  linWp, M0D, M1D, M0P, M1P);
    pad(linB,  linBp, 1, M1D, 1, M1P);
    pad(bnG,   bnGp,  1, M1D, 1, M1P);
    pad(bnB,   bnBp,  1, M1D, 1, M1P);
    pad(fc2W,  fc2Wp, M1D, OUTD, M1P, OUTP);
    pad(fc2B,  fc2Bp, 1, OUTD, 1, OUTP);

    // --- degrees & symmetric norms -----------------------------------------
    fill0_kernel<<<cdiv(2 * NN, T), T, 0, stream>>>(normOut, 2 * NN);  // contiguous pair
    degree_kernel<<<cdiv(EE, T), T, 0, stream>>>(src, dst, normOut, normIn, EE);
    norm_kernel<<<cdiv(NN, T), T, 0, stream>>>(normOut, NN);
    norm_kernel<<<cdiv(NN, T), T, 0, stream>>>(normIn, NN);

    // --- GraphConv layer 1 (38 -> 128, K padded to 40) ----------------------
    fill0_kernel<<<cdiv(NN * F_INP, T), T, 0, stream>>>(buf1, NN * F_INP);
    scatter_kernel<<<cdiv(EE * F_IN, T), T, 0, stream>>>(h_in, src, dst, normOut,
                                                         buf1, EE * F_IN, F_IN, F_IN, F_INP);
    gemm(buf1, W0p, gcnB[0], normIn, buf0, NN, F_INP, HID);
    bn(buf0, gcnG[0], gcnBeta[0], NN, HID);

    // --- GraphConv layers 2,3 (128 -> 128) ---------------------------------
    for (int l = 1; l < 3; ++l) {
        fill0_kernel<<<cdiv(NN * HID, T), T, 0, stream>>>(buf1, NN * HID);
        scatter_kernel<<<cdiv(EE * HID, T), T, 0, stream>>>(buf0, src, dst, normOut,
                                                            buf1, EE * HID, HID, HID, HID);
        gemm(buf1, gcnW[l], gcnB[l], normIn, buf0, NN, HID, HID);
        bn(buf0, gcnG[l], gcnBeta[l], NN, HID);
    }

    // --- WeightAndSum readout ----------------------------------------------
    atom_weight_kernel<<<cdiv(NN, T), T, 0, stream>>>(buf0, rwW, rwB, outAW, sigw, NN, HID);
    fill0_kernel<<<cdiv(GG * HID, T), T, 0, stream>>>(hg, GG * HID);
    readout_scatter_kernel<<<cdiv(NN * HID, T), T, 0, stream>>>(buf0, sigw, gid, hg, NN, HID);

    // --- MLP head (padded dims: 128 -> 512 -> 112 -> 80) --------------------
    gemm(hg, fc1Wp, fc1Bp, nullptr, mbuf0, GG, HID, M0P);
    bn(mbuf0, bn1Gp, bn1Bp, GG, M0P);
    gemm(mbuf0, linWp, linBp, nullptr, mbuf1, GG, M0P, M1P);
    bn(mbuf1, bnGp, bnBp, GG, M1P);
    gemm(mbuf1, fc2Wp, fc2Bp, nullptr, mbuf2, GG, M1P, OUTP);
    sigmoid_slice_kernel<<<cdiv(GG * OUTD, T), T, 0, stream>>>(mbuf2, out, GG, OUTD, OUTP);

    (void)in_sizes; (void)n_in; (void)out_size; (void)ws_size;
}